// BSplineScheduler_85246510891508
// MI455X (gfx1250) — compile-verified
//
#include <hip/hip_runtime.h>
#include <math.h>

// ---------------------------------------------------------------------------
// Monotone cubic B-spline evaluation (N_COEFF=32, ORDER=4, N_TOTAL=34).
// HBM-bound: 32 MB traffic -> ~1.4 us floor @ 23.3 TB/s on MI455X.
// Strategy: per-span power-basis LUT (31 x float4, built once in f64),
// main kernel = float4 stream + LDS b128 gather + Horner (3 fma).
// ---------------------------------------------------------------------------

#define NCOEFF 32
#define NTOTAL 34
#define NSPANS 31   // 31 nonempty knot spans, j in [0,30]

typedef int v4i_t __attribute__((ext_vector_type(4)));

// ===========================================================================
// Kernel 1: build the per-span cubic polynomial table.
// One wave (32 lanes). All lanes redundantly compute the coefficient vector
// (wave-uniform, trivial); lane j < 31 builds span j's polynomial in the
// local coordinate u = 31*s - j (O(1) coefficients => f32-stable table).
// ===========================================================================
__global__ void __launch_bounds__(32)
bspline_build_lut(const float* __restrict__ theta, float4* __restrict__ lut)
{
    const int j = threadIdx.x;

    // --- coefficients: softplus -> cumsum -> normalize -> [0, interior, 1]
    float coeff[NTOTAL];
    float cs[NCOEFF];
    float acc = 0.0f;
    for (int k = 0; k < NCOEFF; ++k) {
        float x  = theta[k];
        // jax.nn.softplus == logaddexp(x, 0) == max(x,0) + log1p(exp(-|x|))
        float sp = fmaxf(x, 0.0f) + log1pf(expf(-fabsf(x)));
        acc += sp;
        cs[k] = acc;
    }
    coeff[0] = 0.0f;
    for (int k = 0; k < NCOEFF; ++k) coeff[k + 1] = cs[k] / acc;
    coeff[NTOTAL - 1] = 1.0f;

    if (j >= NSPANS) return;

    // --- knot vector: kn[m] = clamp(m-3, 0, 31) / 31  (clamped cubic, uniform)
    const double h = 1.0 / 31.0;
    auto kn = [&](int m) -> double {
        int q = m - 3;
        q = q < 0 ? 0 : (q > 31 ? 31 : q);
        return (double)q * h;
    };

    // --- Cox-de Boor on polynomials in u (s = (j+u)*h), f64.
    // N[r][k] = coefficient of u^k of basis function B_{j+r} restricted to span j.
    const int i = j + 3;              // span index in full knot vector
    double N[4][4] = {};
    N[0][0] = 1.0;
    for (int p = 1; p <= 3; ++p) {
        double saved[4] = {0.0, 0.0, 0.0, 0.0};
        for (int r = 0; r < p; ++r) {
            // denominator right[r+1]+left[p-r] = kn[i+r+1]-kn[i+r+1-p] (s cancels)
            const double invd = 1.0 / (kn(i + r + 1) - kn(i + r + 1 - p));
            // right_{r+1}(u) = kn[i+r+1] - s(u) = (kn[i+r+1]-j*h) + (-h)*u
            const double ra0 = kn(i + r + 1) - (double)j * h, ra1 = -h;
            // left_{p-r}(u) = s(u) - kn[i+1-p+r] = (j*h-kn[i+1-p+r]) + h*u
            const double la0 = (double)j * h - kn(i + 1 - p + r), la1 = h;

            double T[4];
            for (int k = 0; k < 4; ++k) T[k] = N[r][k] * invd;

            double n0 = saved[0] + ra0 * T[0];
            double n1 = saved[1] + ra0 * T[1] + ra1 * T[0];
            double n2 = saved[2] + ra0 * T[2] + ra1 * T[1];
            double n3 = saved[3] + ra0 * T[3] + ra1 * T[2];
            N[r][0] = n0; N[r][1] = n1; N[r][2] = n2; N[r][3] = n3;

            saved[0] = la0 * T[0];
            saved[1] = la0 * T[1] + la1 * T[0];
            saved[2] = la0 * T[2] + la1 * T[1];
            saved[3] = la0 * T[3] + la1 * T[2];
        }
        for (int k = 0; k < 4; ++k) N[p][k] = saved[k];
    }

    // spline|span_j (u) = sum_r coeff[j+r] * N_r(u)
    double poly[4] = {0.0, 0.0, 0.0, 0.0};
    for (int r = 0; r < 4; ++r) {
        const double c = (double)coeff[j + r];
        for (int k = 0; k < 4; ++k) poly[k] += N[r][k] * c;
    }
    lut[j] = make_float4((float)poly[0], (float)poly[1],
                         (float)poly[2], (float)poly[3]);
}

// ===========================================================================
// Kernel 2: evaluate. float4 stream; LUT staged to LDS via the gfx1250
// async-to-LDS path; per-element ds_load_b128 gather + Horner.
// ===========================================================================
__device__ __forceinline__ float eval_one(float sv, const float4* lut)
{
    float sc = fminf(fmaxf(sv, 0.0f), 1.0f);
    float t  = sc * 31.0f;
    int   j  = (int)t;
    j = j > (NSPANS - 1) ? (NSPANS - 1) : j;
    float u  = t - (float)j;              // local coordinate in [0,1]
    float4 p = lut[j];                    // ds_load_b128, 16B aligned
    float r  = fmaf(fmaf(fmaf(p.w, u, p.z), u, p.y), u, p.x);
    r = (sc <= 1e-7f) ? 0.0f : r;
    r = (sc >= 1.0f - 1e-7f) ? 1.0f : r;  // 1-override wins, matching reference
    return r;
}

__global__ void __launch_bounds__(256)
bspline_eval(const float4* __restrict__ s4,
             const float*  __restrict__ s,
             const float4* __restrict__ lutg,
             float4* __restrict__ out4,
             float*  __restrict__ out,
             int nvec, int n)
{
    __shared__ float4 lut[NSPANS + 1];
    const int tid = threadIdx.x;

#if defined(__gfx1250__) && __has_builtin(__builtin_amdgcn_global_load_async_to_lds_b128)
    // CDNA5 async-tensor path: 31 lanes each DMA one float4 into LDS.
    if (tid < NSPANS) {
        __builtin_amdgcn_global_load_async_to_lds_b128(
            (__attribute__((address_space(1))) v4i_t*)(lutg + tid),
            (__attribute__((address_space(3))) v4i_t*)(&lut[tid]),
            /*offset=*/0, /*cpol=*/0);
    }
  #if __has_builtin(__builtin_amdgcn_s_wait_asynccnt)
    __builtin_amdgcn_s_wait_asynccnt(0);
  #else
    asm volatile("s_wait_asynccnt 0" ::: "memory");
  #endif
#else
    if (tid < NSPANS) lut[tid] = lutg[tid];
#endif
    __syncthreads();

    const int gid = blockIdx.x * blockDim.x + tid;
    if (gid < nvec) {
        float4 sv = s4[gid];
        float4 r;
        r.x = eval_one(sv.x, lut);
        r.y = eval_one(sv.y, lut);
        r.z = eval_one(sv.z, lut);
        r.w = eval_one(sv.w, lut);
        out4[gid] = r;
    }

    // scalar tail (n % 4 elements); n = 4M in practice => no-op
    if (gid == 0) {
        for (int idx = nvec * 4; idx < n; ++idx)
            out[idx] = eval_one(s[idx], lut);
    }
}

// ===========================================================================
// Launch
// ===========================================================================
extern "C" void kernel_launch(void* const* d_in, const int* in_sizes, int n_in,
                              void* d_out, int out_size, void* d_ws, size_t ws_size,
                              hipStream_t stream)
{
    const float* d_s     = (const float*)d_in[0];   // 's'     [M] f32
    const float* d_theta = (const float*)d_in[1];   // 'theta' [32] f32
    float*       d_o     = (float*)d_out;           // [M] f32
    float4*      d_lut   = (float4*)d_ws;           // 31 * 16 B = 496 B scratch

    const int n = in_sizes[0];

    bspline_build_lut<<<1, 32, 0, stream>>>(d_theta, d_lut);

    const int nvec   = n >> 2;
    int       blocks = (nvec + 255) / 256;
    if (blocks < 1) blocks = 1;
    bspline_eval<<<blocks, 256, 0, stream>>>((const float4*)d_s, d_s, d_lut,
                                             (float4*)d_o, d_o, nvec, n);
}